// GIN_block_40029095198815
// MI455X (gfx1250) — compile-verified
//
#include <hip/hip_runtime.h>

typedef __attribute__((ext_vector_type(2))) float v2f;
typedef __attribute__((ext_vector_type(4))) float v4f;
typedef __attribute__((ext_vector_type(8))) float v8f;

#define N_NODES 50000
#define N_EDGES 800000
#define D 128
#define LDSW 132   // padded row: 132 % 64-bank pattern -> conflict-free b64 reads

// ---------------------------------------------------------------------------
// Kernel 1: zero the aggregation workspace (agg lives in d_ws, 25.6 MB).
// ---------------------------------------------------------------------------
__global__ void zero_agg_kernel(float* __restrict__ agg) {
    int i = blockIdx.x * blockDim.x + threadIdx.x;   // one float4 per thread
    v4f z = {0.f, 0.f, 0.f, 0.f};
    ((v4f*)agg)[i] = z;
}

// ---------------------------------------------------------------------------
// Kernel 2: edge scatter-add. One wave32 per edge; each lane handles 4 feats.
// agg (25.6 MB) is L2-resident on MI455X (192 MB L2) -> atomics stay in L2.
// ---------------------------------------------------------------------------
__global__ void scatter_kernel(const float* __restrict__ x,
                               const int* __restrict__ src,
                               const int* __restrict__ dst,
                               float* __restrict__ agg) {
    int t = blockIdx.x * blockDim.x + threadIdx.x;
    int e = t >> 5;
    int lane = t & 31;
    if (e >= N_EDGES) return;
    int s = src[e];
    int d = dst[e];
    const v4f xv = *(const v4f*)(x + (size_t)s * D + lane * 4);
    float* ap = agg + (size_t)d * D + lane * 4;
    atomicAdd(ap + 0, xv.x);
    atomicAdd(ap + 1, xv.y);
    atomicAdd(ap + 2, xv.z);
    atomicAdd(ap + 3, xv.w);
}

// ---------------------------------------------------------------------------
// Kernel 3: fused GIN MLP + residual using V_WMMA_F32_16X16X4_F32.
//   out = relu((x+agg)@W1 + b1) @ W2 + b2 + x@Wres + bres
// Block = 16 output rows x 128 cols; 8 waves, wave w owns 16-col N-tile w.
//
// f32 WMMA fragment layouts (wave32):
//   A 16x4:  lane l -> M = l%16;  VGPR v holds K = v + 2*(l/16)
//            => {V0,V1} are consecutive K values -> single b64 LDS load.
//   B 4x16:  lane l -> N = l%16;  VGPR v holds K-row v + 2*(l/16)  (symmetric)
//   C/D 16x16: lane l -> N = l%16; VGPR r holds M = r + 8*(l/16)
// ---------------------------------------------------------------------------
__global__ __launch_bounds__(256) void gin_wmma_kernel(
    const float* __restrict__ x,   const float* __restrict__ agg,
    const float* __restrict__ W1,  const float* __restrict__ b1,
    const float* __restrict__ W2,  const float* __restrict__ b2,
    const float* __restrict__ Wres,const float* __restrict__ bres,
    float* __restrict__ out) {

    __shared__ float sh_xa[16 * LDSW];  // x + agg    (GIN input)
    __shared__ float sh_xr[16 * LDSW];  // x          (residual input)
    __shared__ float sh_t1[16 * LDSW];  // relu(xa@W1 + b1)

    const int tid = threadIdx.x;
    const int rowbase = blockIdx.x * 16;   // 50000 / 16 = 3125 blocks exactly

    // Cooperative tile load: 16 rows x 128 cols, as float4 (512 float4 total).
    for (int i = tid; i < 16 * (D / 4); i += 256) {
        int r  = i >> 5;        // D/4 == 32
        int c4 = (i & 31) * 4;
        v4f xv = *(const v4f*)(x   + (size_t)(rowbase + r) * D + c4);
        v4f av = *(const v4f*)(agg + (size_t)(rowbase + r) * D + c4);
        v4f sv = {xv.x + av.x, xv.y + av.y, xv.z + av.z, xv.w + av.w};
        *(v4f*)(sh_xr + r * LDSW + c4) = xv;
        *(v4f*)(sh_xa + r * LDSW + c4) = sv;
    }
    __syncthreads();

    const int lane = tid & 31;      // wave32
    const int wv   = tid >> 5;      // 8 waves
    const int n    = lane & 15;     // N (and M for A-fragment) index
    const int h    = lane >> 4;     // lane half
    const int colbase = wv * 16;

    // ---- Stage 1: T1 = relu(xa @ W1 + b1), 16x16 tile per wave -------------
    v8f acc = {};
    #pragma unroll
    for (int k = 0; k < D; k += 4) {
        v2f a = *(const v2f*)(sh_xa + n * LDSW + k + 2 * h);   // b64, conflict-free
        v2f b;
        b.x = W1[(size_t)(k + 2 * h + 0) * D + colbase + n];
        b.y = W1[(size_t)(k + 2 * h + 1) * D + colbase + n];
        acc = __builtin_amdgcn_wmma_f32_16x16x4_f32(
                  false, a, false, b, (short)0, acc, false, false);
    }
    const float bias1 = b1[colbase + n];
    #pragma unroll
    for (int r = 0; r < 8; ++r) {
        float t = acc[r] + bias1;
        sh_t1[(r + 8 * h) * LDSW + colbase + n] = fmaxf(t, 0.f);
    }
    __syncthreads();

    // ---- Stage 2: out = T1@W2 + x@Wres + (b2 + bres), fused accumulator ----
    v8f acc2 = {};
    #pragma unroll
    for (int k = 0; k < D; k += 4) {
        v2f a1 = *(const v2f*)(sh_t1 + n * LDSW + k + 2 * h);
        v2f bw2;
        bw2.x = W2[(size_t)(k + 2 * h + 0) * D + colbase + n];
        bw2.y = W2[(size_t)(k + 2 * h + 1) * D + colbase + n];
        acc2 = __builtin_amdgcn_wmma_f32_16x16x4_f32(
                   false, a1, false, bw2, (short)0, acc2, false, false);

        v2f a2 = *(const v2f*)(sh_xr + n * LDSW + k + 2 * h);
        v2f bwr;
        bwr.x = Wres[(size_t)(k + 2 * h + 0) * D + colbase + n];
        bwr.y = Wres[(size_t)(k + 2 * h + 1) * D + colbase + n];
        acc2 = __builtin_amdgcn_wmma_f32_16x16x4_f32(
                   false, a2, false, bwr, (short)0, acc2, false, false);
    }
    const float bias2 = b2[colbase + n] + bres[colbase + n];
    #pragma unroll
    for (int r = 0; r < 8; ++r) {
        out[(size_t)(rowbase + r + 8 * h) * D + colbase + n] = acc2[r] + bias2;
    }
}

// ---------------------------------------------------------------------------
// Launch. Input order: x, edge_index, W1, b1, W2, b2, Wres, bres.
// d_ws: agg accumulator (N_NODES * D floats = 25.6 MB).
// ---------------------------------------------------------------------------
extern "C" void kernel_launch(void* const* d_in, const int* in_sizes, int n_in,
                              void* d_out, int out_size, void* d_ws, size_t ws_size,
                              hipStream_t stream) {
    const float* x    = (const float*)d_in[0];
    const int*   ei   = (const int*)  d_in[1];   // [2, N_EDGES]: row0=src, row1=dst
    const float* W1   = (const float*)d_in[2];
    const float* b1   = (const float*)d_in[3];
    const float* W2   = (const float*)d_in[4];
    const float* b2   = (const float*)d_in[5];
    const float* Wres = (const float*)d_in[6];
    const float* bres = (const float*)d_in[7];
    float* out = (float*)d_out;
    float* agg = (float*)d_ws;

    // 1) agg = 0                  (50000*128/4 float4 / 256 = 6250 blocks)
    zero_agg_kernel<<<(N_NODES * D / 4) / 256, 256, 0, stream>>>(agg);

    // 2) agg[dst] += x[src]       (one wave per edge: 800000*32/256 = 100000)
    scatter_kernel<<<(N_EDGES * 32) / 256, 256, 0, stream>>>(
        x, ei, ei + N_EDGES, agg);

    // 3) fused WMMA MLP + residual (50000/16 = 3125 blocks)
    gin_wmma_kernel<<<N_NODES / 16, 256, 0, stream>>>(
        x, agg, W1, b1, W2, b2, Wres, bres, out);
}